// uncertainty_estimator_85255100825936
// MI455X (gfx1250) — compile-verified
//
#include <hip/hip_runtime.h>

#define N_NODES 100000
#define N_EDGES 3200000
#define FEAT 128   // FEAT == HID == OUT == 128

typedef float v2f __attribute__((ext_vector_type(2)));
typedef float v8f __attribute__((ext_vector_type(8)));

// Hardware f32 atomic add (global_atomic_add_f32), never a CAS loop.
__device__ __forceinline__ void atomAddF(float* p, float v) {
    unsafeAtomicAdd(p, v);
}

// ---------------------------------------------------------------------------
// zero fill (n must be a multiple of 4)
// ---------------------------------------------------------------------------
__global__ void zero_f32(float* __restrict__ p, long n) {
    long n4 = n >> 2;
    long stride = (long)gridDim.x * blockDim.x;
    for (long i = (long)blockIdx.x * blockDim.x + threadIdx.x; i < n4; i += stride) {
        ((float4*)p)[i] = make_float4(0.f, 0.f, 0.f, 0.f);
    }
}

// ---------------------------------------------------------------------------
// degree accumulation: deg[dst] += 1
// ---------------------------------------------------------------------------
__global__ void degree_kernel(const long long* __restrict__ ei, float* __restrict__ deg,
                              int nEdges) {
    int e = blockIdx.x * blockDim.x + threadIdx.x;
    if (e < nEdges) {
        int d = (int)ei[(size_t)nEdges + e];
        atomAddF(&deg[d], 1.0f);
    }
}

// dinv = rsqrt(deg + 1)   (self loop)
__global__ void dinv_kernel(const float* __restrict__ deg, float* __restrict__ dinv, int n) {
    int i = blockIdx.x * blockDim.x + threadIdx.x;
    if (i < n) dinv[i] = rsqrtf(deg[i] + 1.0f);
}

// ---------------------------------------------------------------------------
// WMMA f32 GEMM:  C[M x 128] = A[M x 128] * W[128 x 128]  (+ optional bias)
// blockDim = 128 (4 waves). Each block does a 16-row stripe; each wave does
// two 16x16 output tiles. A tile staged in LDS (padded 132 to avoid the
// 16-way bank conflict on column reads), W streamed from global (hot in L2).
// K loop fully unrolled: 32 chained v_wmma_f32_16x16x4_f32 per tile.
// ---------------------------------------------------------------------------
__global__ void gemm128_wmma(const float* __restrict__ A, const float* __restrict__ W,
                             const float* __restrict__ bias, float* __restrict__ C) {
    __shared__ float As[16 * 132];
    const int m0 = blockIdx.x * 16;
    const int t  = threadIdx.x;

    // cooperative load: 16 rows x 128 cols = 512 float4 across 128 threads
    for (int i = t; i < 16 * 32; i += 128) {
        int r  = i >> 5;
        int c4 = i & 31;
        float4 v = ((const float4*)(A + (size_t)(m0 + r) * FEAT))[c4];
        float* dst = &As[r * 132 + c4 * 4];
        dst[0] = v.x; dst[1] = v.y; dst[2] = v.z; dst[3] = v.w;
    }
    __syncthreads();

    const int wave = t >> 5;
    const int lane = t & 31;
    const int row  = lane & 15;   // M index within tile (A), N index (B/C)
    const int half = lane >> 4;   // selects K pair / upper 8 rows of C

    #pragma unroll
    for (int tt = 0; tt < 2; ++tt) {
        const int n0 = (wave * 2 + tt) * 16;
        v8f acc = {};
        #pragma unroll
        for (int kk = 0; kk < 128; kk += 4) {
            const int k = kk + 2 * half;
            // A 16x4 layout: lanes 0-15 rows, v0 = K={0|2}, v1 = K={1|3}
            v2f a;
            a.x = As[row * 132 + k];
            a.y = As[row * 132 + k + 1];
            // B 4x16 layout (mirrored): lanes hold N, halves hold K pairs
            v2f b;
            b.x = W[(size_t)k * FEAT + n0 + row];
            b.y = W[(size_t)(k + 1) * FEAT + n0 + row];
            acc = __builtin_amdgcn_wmma_f32_16x16x4_f32(
                /*neg_a=*/false, a, /*neg_b=*/false, b,
                /*c_mod=*/(short)0, acc, /*reuse_a=*/false, /*reuse_b=*/false);
        }
        const float bv = bias ? bias[n0 + row] : 0.0f;
        // C/D layout: VGPR r -> M = r + 8*half, N = lane&15
        #pragma unroll
        for (int r = 0; r < 8; ++r) {
            const int mrow = m0 + r + 8 * half;
            C[(size_t)mrow * FEAT + n0 + row] = acc[r] + bv;
        }
    }
}

// ---------------------------------------------------------------------------
// Edge aggregation: one wave per edge, 4 features per lane.
// agg[dst] += h[src] * dinv[src]*dinv[dst]  via hardware f32 atomics (L2).
// Next edge's source row is prefetched (global_prefetch_b8) to hide the
// random-gather latency; dst buffer (51 MB) is L2-resident so the atomics
// run at L2 bandwidth.
// ---------------------------------------------------------------------------
__global__ void aggregate_kernel(const float* __restrict__ h, const long long* __restrict__ ei,
                                 const float* __restrict__ dinv, float* __restrict__ agg,
                                 int nEdges) {
    const int lane   = threadIdx.x & 31;
    const int wave   = (blockIdx.x * blockDim.x + threadIdx.x) >> 5;
    const int nWaves = (gridDim.x * blockDim.x) >> 5;
    for (int e = wave; e < nEdges; e += nWaves) {
        const int s = (int)ei[e];
        const int d = (int)ei[(size_t)nEdges + e];

        // prefetch next edge's gather row into cache
        const int e2 = e + nWaves;
        if (e2 < nEdges) {
            const int s2 = (int)ei[e2];
            __builtin_prefetch(h + (size_t)s2 * FEAT + lane * 4, 0, 0);
        }

        const float nrm = dinv[s] * dinv[d];
        const float4 hv = ((const float4*)(h + (size_t)s * FEAT))[lane];
        float* ap = agg + (size_t)d * FEAT + lane * 4;
        atomAddF(ap + 0, hv.x * nrm);
        atomAddF(ap + 1, hv.y * nrm);
        atomAddF(ap + 2, hv.z * nrm);
        atomAddF(ap + 3, hv.w * nrm);
    }
}

// ---------------------------------------------------------------------------
// out = relu(agg + h * dinv^2 + b)   (self-loop + bias + ReLU, fused)
// one thread per float4; 32 threads per node row.
// ---------------------------------------------------------------------------
__global__ void post_relu_kernel(const float* __restrict__ agg, const float* __restrict__ h,
                                 const float* __restrict__ dinv, const float* __restrict__ b,
                                 float* __restrict__ out, int nNodes) {
    const long tid = (long)blockIdx.x * blockDim.x + threadIdx.x;
    if (tid >= (long)nNodes * 32) return;
    const int node = (int)(tid >> 5);
    const int q    = (int)(tid & 31);
    const float di = dinv[node];
    const float sc = di * di;
    const size_t idx = (size_t)node * 32 + q;
    const float4 hv = ((const float4*)h)[idx];
    const float4 av = ((const float4*)agg)[idx];
    const float4 bv = ((const float4*)b)[q];
    float4 r;
    r.x = fmaxf(fmaf(hv.x, sc, av.x) + bv.x, 0.0f);
    r.y = fmaxf(fmaf(hv.y, sc, av.y) + bv.y, 0.0f);
    r.z = fmaxf(fmaf(hv.z, sc, av.z) + bv.z, 0.0f);
    r.w = fmaxf(fmaf(hv.w, sc, av.w) + bv.w, 0.0f);
    ((float4*)out)[idx] = r;
}

// ---------------------------------------------------------------------------
extern "C" void kernel_launch(void* const* d_in, const int* in_sizes, int n_in,
                              void* d_out, int out_size, void* d_ws, size_t ws_size,
                              hipStream_t stream) {
    const float*     x  = (const float*)d_in[0];
    const long long* ei = (const long long*)d_in[1];   // int64 edge_index [2, E]
    const float*     W1 = (const float*)d_in[2];
    const float*     b1 = (const float*)d_in[3];
    const float*     W2 = (const float*)d_in[4];
    const float*     b2 = (const float*)d_in[5];
    const float*     Wl = (const float*)d_in[6];
    const float*     bl = (const float*)d_in[7];
    float* out = (float*)d_out;

    // workspace partition: 2 feature buffers + deg + dinv (~103 MB)
    float* BUF0 = (float*)d_ws;                         // [N, 128]
    float* BUF1 = BUF0 + (size_t)N_NODES * FEAT;        // [N, 128]
    float* DEG  = BUF1 + (size_t)N_NODES * FEAT;        // [N]
    float* DINV = DEG + N_NODES;                        // [N]

    const int ZB   = 4096;   // blocks for grid-stride zero / aggregate
    const int EB   = (N_EDGES + 255) / 256;
    const int NB   = (N_NODES + 255) / 256;
    const int PB   = ((int)((long)N_NODES * 32 + 255)) / 256;
    const int GEMB = N_NODES / 16;   // 6250

    // degrees (with self loop) -> dinv
    zero_f32<<<256, 256, 0, stream>>>(DEG, N_NODES);
    degree_kernel<<<EB, 256, 0, stream>>>(ei, DEG, N_EDGES);
    dinv_kernel<<<NB, 256, 0, stream>>>(DEG, DINV, N_NODES);

    // ---- layer 1 ----
    gemm128_wmma<<<GEMB, 128, 0, stream>>>(x, W1, nullptr, BUF0);
    zero_f32<<<ZB, 256, 0, stream>>>(BUF1, (long)N_NODES * FEAT);
    aggregate_kernel<<<ZB, 256, 0, stream>>>(BUF0, ei, DINV, BUF1, N_EDGES);
    post_relu_kernel<<<PB, 256, 0, stream>>>(BUF1, BUF0, DINV, b1, out, N_NODES);

    // ---- layer 2 ----
    gemm128_wmma<<<GEMB, 128, 0, stream>>>(out, W2, nullptr, BUF0);
    zero_f32<<<ZB, 256, 0, stream>>>(BUF1, (long)N_NODES * FEAT);
    aggregate_kernel<<<ZB, 256, 0, stream>>>(BUF0, ei, DINV, BUF1, N_EDGES);
    post_relu_kernel<<<PB, 256, 0, stream>>>(BUF1, BUF0, DINV, b2, out, N_NODES);

    // ---- final linear: out = out @ Wl + bl  (in-place; A tile staged in LDS) ----
    gemm128_wmma<<<GEMB, 128, 0, stream>>>(out, Wl, bl, out);
}